// FeaturePropagationLayer_42314017800358
// MI455X (gfx1250) — compile-verified
//
#include <hip/hip_runtime.h>

// ---------------------------------------------------------------------------
// FeaturePropagation (PointNet++ FP layer) for MI455X / gfx1250, wave32.
// Pipeline: 3-NN (VALU) -> gather+mean -> WMMA f32 GEMM1 -> BN stats ->
//           WMMA f32 GEMM2 (BN+ReLU fused on A-load) -> BN stats -> finalize.
// fp32 WMMA (V_WMMA_F32_16X16X4_F32) chosen: workload is HBM-bound (~0.4 GB
// traffic @ 23.3 TB/s dominates the 35 GFLOP of GEMM), so full precision is
// free and matches the fp32 reference exactly.
// ---------------------------------------------------------------------------

typedef float v2f __attribute__((ext_vector_type(2)));
typedef float v8f __attribute__((ext_vector_type(8)));

#define B_    8
#define N_    4096
#define M_    1024
#define C1_   256
#define C2_   512
#define INCH  768            // C1 + C2
#define H1CH  512
#define H2CH  256
#define P_    (B_ * N_)      // 32768 points total
#define L1ROW (3 + C1_)      // 259
#define L2ROW (3 + C2_)      // 515
#define EPS_  1e-5f
#define STATS_G 256          // P_/128 row-groups for column stats

// ------------------------------ 3-NN search --------------------------------
// One block per 256 query points of one batch. xyz2 (+ squared norm) of the
// whole batch is staged in LDS (16 KB). Rank by (||x2||^2 - 2 x1.x2).
__global__ __launch_bounds__(256) void knn3_kernel(const float* __restrict__ l1,
                                                   const float* __restrict__ l2,
                                                   int* __restrict__ idx3) {
    __shared__ float4 s2[M_];
    const int b = blockIdx.y;
    const int n = blockIdx.x * 256 + threadIdx.x;
    for (int j = threadIdx.x; j < M_; j += 256) {
        const float* r = l2 + ((size_t)b * M_ + j) * L2ROW;
        const float x = r[0], y = r[1], z = r[2];
        s2[j] = make_float4(x, y, z, x * x + y * y + z * z);
    }
    __syncthreads();
    const float* r1 = l1 + ((size_t)b * N_ + n) * L1ROW;
    const float x = r1[0], y = r1[1], z = r1[2];
    float d0 = 3.4e38f, d1 = 3.4e38f, d2 = 3.4e38f;
    int   i0 = 0, i1 = 0, i2 = 0;
    for (int m = 0; m < M_; ++m) {
        const float4 q = s2[m];
        const float d = q.w - 2.0f * (x * q.x + y * q.y + z * q.z);
        if (d < d2) {
            if (d < d1) {
                if (d < d0) { d2 = d1; i2 = i1; d1 = d0; i1 = i0; d0 = d; i0 = m; }
                else        { d2 = d1; i2 = i1; d1 = d;  i1 = m; }
            } else          { d2 = d;  i2 = m; }
        }
    }
    int* o = idx3 + ((size_t)b * N_ + n) * 3;
    o[0] = i0; o[1] = i1; o[2] = i2;
}

// --------------------- gather + mean + concat into X -----------------------
// Reference's 1/dist weights collapse to uniform (dist clamped to 1e-10), so
// interpolation == mean of the 3 NN feature rows.
__global__ __launch_bounds__(256) void build_x_kernel(const float* __restrict__ l1,
                                                      const float* __restrict__ l2,
                                                      const int* __restrict__ idx3,
                                                      float* __restrict__ X) {
    const int p = blockIdx.x;
    const int b = p / N_;
    const int t = threadIdx.x;
    const int* id = idx3 + (size_t)p * 3;
    const float* f1 = l1 + (size_t)p * L1ROW + 3;
    const float* fa = l2 + ((size_t)b * M_ + id[0]) * L2ROW + 3;
    const float* fb = l2 + ((size_t)b * M_ + id[1]) * L2ROW + 3;
    const float* fc = l2 + ((size_t)b * M_ + id[2]) * L2ROW + 3;
    float* xr = X + (size_t)p * INCH;
    xr[t] = f1[t];                                   // 256 skip-link channels
#pragma unroll
    for (int cc = 0; cc < 2; ++cc) {                 // 512 interpolated channels
        const int c = t + cc * 256;
        xr[C1_ + c] = (fa[c] + fb[c] + fc[c]) * (1.0f / 3.0f);
    }
}

// ----------------------- WMMA f32 GEMM (+fused BN in) ----------------------
// Out[m][n] = sum_k act(A[m][k]) * W[n][k] + bias[n]
// act = identity, or relu(a*scale[k]+shift[k]) when BN_IN (fused BN+ReLU of
// the previous layer). Block tile 128x64, K-chunk 32, 8 waves; each wave owns
// a 16x64 slab via 4 v_wmma_f32_16x16x4_f32 accumulators.
template <int K, int NOUT, bool BN_IN, int OUT_LD, int OUT_OFF>
__global__ __launch_bounds__(256) void gemm_wmma_kernel(const float* __restrict__ A,
                                                        const float* __restrict__ W,
                                                        const float* __restrict__ bias,
                                                        const float* __restrict__ scale,
                                                        const float* __restrict__ shift,
                                                        float* __restrict__ Out) {
    constexpr int BM = 128, BNT = 64, BK = 32, LDSS = 36;   // 36: bank-conflict-free pad
    __shared__ float Xs[BM * LDSS];
    __shared__ float Wsh[BNT * LDSS];

    const int tid   = threadIdx.x;
    const int wave  = tid >> 5;
    const int lane  = tid & 31;
    const int mrow  = lane & 15;          // fragment row (A) / col (B)
    const int khalf = (lane >> 4) * 2;    // lanes 16..31 carry K pair {2,3}
    const int rowBlock = blockIdx.y * BM;
    const int colBlock = blockIdx.x * BNT;

    v8f acc0 = {}, acc1 = {}, acc2 = {}, acc3 = {};

    for (int kc = 0; kc < K; kc += BK) {
        // Stage A chunk 128x32 (coalesced float4), fusing BN+ReLU if requested.
#pragma unroll
        for (int rnd = 0; rnd < 4; ++rnd) {
            const int r = rnd * 32 + (tid >> 3);
            const int c = (tid & 7) * 4;
            float4 v = *(const float4*)(A + (size_t)(rowBlock + r) * K + kc + c);
            if (BN_IN) {
                v.x = fmaxf(v.x * scale[kc + c + 0] + shift[kc + c + 0], 0.0f);
                v.y = fmaxf(v.y * scale[kc + c + 1] + shift[kc + c + 1], 0.0f);
                v.z = fmaxf(v.z * scale[kc + c + 2] + shift[kc + c + 2], 0.0f);
                v.w = fmaxf(v.w * scale[kc + c + 3] + shift[kc + c + 3], 0.0f);
            }
            Xs[r * LDSS + c + 0] = v.x; Xs[r * LDSS + c + 1] = v.y;
            Xs[r * LDSS + c + 2] = v.z; Xs[r * LDSS + c + 3] = v.w;
        }
        // Stage W chunk 64x32.
#pragma unroll
        for (int rnd = 0; rnd < 2; ++rnd) {
            const int r = rnd * 32 + (tid >> 3);
            const int c = (tid & 7) * 4;
            const float4 v = *(const float4*)(W + (size_t)(colBlock + r) * K + kc + c);
            Wsh[r * LDSS + c + 0] = v.x; Wsh[r * LDSS + c + 1] = v.y;
            Wsh[r * LDSS + c + 2] = v.z; Wsh[r * LDSS + c + 3] = v.w;
        }
        __syncthreads();

#pragma unroll
        for (int k4 = 0; k4 < BK; k4 += 4) {
            const float* ap = &Xs[(wave * 16 + mrow) * LDSS + k4 + khalf];
            v2f a; a.x = ap[0]; a.y = ap[1];
            v2f fb0, fb1, fb2, fb3;
            const float* bp0 = &Wsh[( 0 + mrow) * LDSS + k4 + khalf];
            const float* bp1 = &Wsh[(16 + mrow) * LDSS + k4 + khalf];
            const float* bp2 = &Wsh[(32 + mrow) * LDSS + k4 + khalf];
            const float* bp3 = &Wsh[(48 + mrow) * LDSS + k4 + khalf];
            fb0.x = bp0[0]; fb0.y = bp0[1];
            fb1.x = bp1[0]; fb1.y = bp1[1];
            fb2.x = bp2[0]; fb2.y = bp2[1];
            fb3.x = bp3[0]; fb3.y = bp3[1];
            acc0 = __builtin_amdgcn_wmma_f32_16x16x4_f32(false, a, false, fb0, (short)0, acc0, false, false);
            acc1 = __builtin_amdgcn_wmma_f32_16x16x4_f32(false, a, false, fb1, (short)0, acc1, false, false);
            acc2 = __builtin_amdgcn_wmma_f32_16x16x4_f32(false, a, false, fb2, (short)0, acc2, false, false);
            acc3 = __builtin_amdgcn_wmma_f32_16x16x4_f32(false, a, false, fb3, (short)0, acc3, false, false);
        }
        __syncthreads();
    }

    // Epilogue: D layout => VGPR v holds M = v (lanes 0-15) / v+8 (lanes 16-31).
    const int nb    = lane & 15;
    const int half8 = (lane >> 4) * 8;
    const float bia0 = bias[colBlock +  0 + nb];
    const float bia1 = bias[colBlock + 16 + nb];
    const float bia2 = bias[colBlock + 32 + nb];
    const float bia3 = bias[colBlock + 48 + nb];
#pragma unroll
    for (int v = 0; v < 8; ++v) {
        const size_t m = (size_t)(rowBlock + wave * 16 + half8 + v);
        float* orow = Out + m * OUT_LD + OUT_OFF;
        orow[colBlock +  0 + nb] = acc0[v] + bia0;
        orow[colBlock + 16 + nb] = acc1[v] + bia1;
        orow[colBlock + 32 + nb] = acc2[v] + bia2;
        orow[colBlock + 48 + nb] = acc3[v] + bia3;
    }
}

// ----------------- deterministic per-channel mean/var stats ----------------
template <int C, int LD, int OFF>
__global__ __launch_bounds__(256) void colstats_partial_kernel(const float* __restrict__ H,
                                                               float* __restrict__ psum,
                                                               float* __restrict__ psq) {
    const int g = blockIdx.x;   // row-group of 128 points
    const int t = threadIdx.x;
#pragma unroll
    for (int cc = 0; cc < C / 256; ++cc) {
        const int c = t + cc * 256;
        float s = 0.0f, q = 0.0f;
        for (int r = 0; r < 128; ++r) {
            const float v = H[(size_t)(g * 128 + r) * LD + OFF + c];
            s += v;
            q += v * v;
        }
        psum[(size_t)g * C + c] = s;
        psq [(size_t)g * C + c] = q;
    }
}

template <int C>
__global__ void colstats_finalize_kernel(const float* __restrict__ psum,
                                         const float* __restrict__ psq,
                                         const float* __restrict__ gamma,
                                         const float* __restrict__ beta,
                                         float* __restrict__ scale,
                                         float* __restrict__ shift) {
    const int c = threadIdx.x;
    if (c >= C) return;
    float s = 0.0f, q = 0.0f;
    for (int g = 0; g < STATS_G; ++g) {
        s += psum[(size_t)g * C + c];
        q += psq [(size_t)g * C + c];
    }
    const float inv  = 1.0f / (float)P_;
    const float mu   = s * inv;
    const float var  = q * inv - mu * mu;
    const float rstd = rsqrtf(var + EPS_);
    const float sc   = gamma[c] * rstd;
    scale[c] = sc;
    shift[c] = beta[c] - mu * sc;
}

// ------------------- final BN2+ReLU in place + xyz copy --------------------
__global__ __launch_bounds__(256) void out_finalize_kernel(const float* __restrict__ l1,
                                                           const float* __restrict__ scale,
                                                           const float* __restrict__ shift,
                                                           float* __restrict__ Out) {
    const int p = blockIdx.x;
    const int t = threadIdx.x;   // 256 == H2CH
    float* orow = Out + (size_t)p * L1ROW;
    const float raw = orow[3 + t];
    orow[3 + t] = fmaxf(raw * scale[t] + shift[t], 0.0f);
    if (t < 3) orow[t] = l1[(size_t)p * L1ROW + t];
}

// ---------------------------------------------------------------------------
extern "C" void kernel_launch(void* const* d_in, const int* in_sizes, int n_in,
                              void* d_out, int out_size, void* d_ws, size_t ws_size,
                              hipStream_t stream) {
    const float* l1  = (const float*)d_in[0];
    const float* l2  = (const float*)d_in[1];
    const float* W1  = (const float*)d_in[2];
    const float* b1  = (const float*)d_in[3];
    const float* g1  = (const float*)d_in[4];
    const float* be1 = (const float*)d_in[5];
    const float* W2  = (const float*)d_in[6];
    const float* b2  = (const float*)d_in[7];
    const float* g2  = (const float*)d_in[8];
    const float* be2 = (const float*)d_in[9];
    float* out = (float*)d_out;

    char* wsb = (char*)d_ws;
    size_t off = 0;
    auto alloc = [&](size_t bytes) -> void* {
        void* p = wsb + off;
        off = (off + bytes + 255) & ~(size_t)255;
        return p;
    };
    float* X   = (float*)alloc((size_t)P_ * INCH * sizeof(float));   // ~100.7 MB
    float* H1  = (float*)alloc((size_t)P_ * H1CH * sizeof(float));   //  ~67.1 MB
    int*   idx = (int*)  alloc((size_t)P_ * 3 * sizeof(int));
    float* ps1 = (float*)alloc((size_t)STATS_G * H1CH * sizeof(float));
    float* pq1 = (float*)alloc((size_t)STATS_G * H1CH * sizeof(float));
    float* ps2 = (float*)alloc((size_t)STATS_G * H2CH * sizeof(float));
    float* pq2 = (float*)alloc((size_t)STATS_G * H2CH * sizeof(float));
    float* sc1 = (float*)alloc(H1CH * sizeof(float));
    float* sh1 = (float*)alloc(H1CH * sizeof(float));
    float* sc2 = (float*)alloc(H2CH * sizeof(float));
    float* sh2 = (float*)alloc(H2CH * sizeof(float));

    knn3_kernel<<<dim3(N_ / 256, B_), 256, 0, stream>>>(l1, l2, idx);
    build_x_kernel<<<P_, 256, 0, stream>>>(l1, l2, idx, X);

    // Layer 1: H1 = X @ W1^T + b1  (raw, pre-BN)
    gemm_wmma_kernel<INCH, H1CH, false, H1CH, 0>
        <<<dim3(H1CH / 64, P_ / 128), 256, 0, stream>>>(X, W1, b1, nullptr, nullptr, H1);
    colstats_partial_kernel<H1CH, H1CH, 0><<<STATS_G, 256, 0, stream>>>(H1, ps1, pq1);
    colstats_finalize_kernel<H1CH><<<1, H1CH, 0, stream>>>(ps1, pq1, g1, be1, sc1, sh1);

    // Layer 2: out.feat = relu(BN1(H1)) @ W2^T + b2  (BN1+ReLU fused on load)
    gemm_wmma_kernel<H1CH, H2CH, true, L1ROW, 3>
        <<<dim3(H2CH / 64, P_ / 128), 256, 0, stream>>>(H1, W2, b2, sc1, sh1, out);
    colstats_partial_kernel<H2CH, L1ROW, 3><<<STATS_G, 256, 0, stream>>>(out, ps2, pq2);
    colstats_finalize_kernel<H2CH><<<1, H2CH, 0, stream>>>(ps2, pq2, g2, be2, sc2, sh2);

    out_finalize_kernel<<<P_, 256, 0, stream>>>(l1, sc2, sh2, out);
}